// CausalSelfAttention_13073880449318
// MI455X (gfx1250) — compile-verified
//
#include <hip/hip_runtime.h>
#include <hip/hip_bf16.h>

// ---------------------------------------------------------------------------
// CausalSelfAttention for MI455X (gfx1250), fp32 end-to-end via
// V_WMMA_F32_16X16X4_F32.  Memory-bound: the 1 GiB attn_weights output write
// dominates (~46us floor at 23.3 TB/s), so fp32 matrix math is free and the
// priority is wide (b128) vectorized + asynchronous data movement.
// ---------------------------------------------------------------------------

typedef __attribute__((ext_vector_type(2))) float v2f;
typedef __attribute__((ext_vector_type(8))) float v8f;
typedef __attribute__((ext_vector_type(4))) int   v4i;

// D = A(16x4,f32) * B(4x16,f32) + C(16x16,f32)
static __device__ __forceinline__ v8f wmma_f32(v2f a, v2f b, v8f c) {
    return __builtin_amdgcn_wmma_f32_16x16x4_f32(false, a, false, b,
                                                 (short)0, c, false, false);
}

// ---- async global->LDS b128 copy (GLOBAL_LOAD_ASYNC_TO_LDS_B128) ----------
#if __has_builtin(__builtin_amdgcn_global_load_async_to_lds_b128) && \
    __has_builtin(__builtin_amdgcn_s_wait_asynccnt)
#define HAS_ASYNC_LDS 1
#else
#define HAS_ASYNC_LDS 0
#endif

typedef v4i __attribute__((address_space(1))) gv4i;   // global int4
typedef v4i __attribute__((address_space(3))) lv4i;   // LDS int4

static __device__ __forceinline__ void cp_async_b128(const float* g, float* l) {
#if HAS_ASYNC_LDS
    __builtin_amdgcn_global_load_async_to_lds_b128(
        (gv4i*)g, (lv4i*)l, 0, 0);
#else
    *(float4*)l = *(const float4*)g;
#endif
}
static __device__ __forceinline__ void cp_async_wait() {
#if HAS_ASYNC_LDS
    __builtin_amdgcn_s_wait_asynccnt(0);
#endif
}

#define D_MODEL 1024
#define N_HEADS 16
#define D_HEAD  64
#define T_SEQ   2048
#define B_SZ    4
#define ROWS    (B_SZ * T_SEQ)      // 8192
#define QKV_M   (3 * D_MODEL)       // 3072

// ---------------------------------------------------------------------------
// GEMM: Y[R, M] = X[R, K] @ W[M, K]^T + bias[M]
// block = 256 threads (8 wave32); macro-tile 32 rows x 64 cols; each wave
// owns one 16x16 tile; K chunked by 32, double-buffered in LDS via async
// global->LDS b128 copies (issue chunk i+1, compute chunk i).
// LDS pitch 36: rows 16B-aligned, 36*r mod 64 distinct for r=0..15 -> no
// bank conflicts on column-wise reads.
// grid: (M/64, R/32)
// ---------------------------------------------------------------------------
#define GP 36
__global__ __launch_bounds__(256)
void gemm_xwt_bias_kernel(const float* __restrict__ X,
                          const float* __restrict__ W,
                          const float* __restrict__ bias,
                          float* __restrict__ Y,
                          int K, int M) {
    __shared__ float Xs[2][32 * GP];
    __shared__ float Ws[2][64 * GP];

    const int tid  = threadIdx.x;
    const int lane = tid & 31;
    const int wave = tid >> 5;

    const int rowBase = blockIdx.y * 32;
    const int colBase = blockIdx.x * 64;
    const int tr = (wave >> 2) * 16;     // 0 / 16   (wave's tile row)
    const int tc = (wave & 3) * 16;      // 0..48    (wave's tile col)

    const int m  = lane & 15;            // A row / B col handled by this lane
    const int kb = (lane < 16) ? 0 : 2;  // K sub-offset per ISA layout

    // per-thread staging coordinates
    const int xr = tid >> 3, xq = (tid & 7) << 2;        // X: 1 float4/thread
    const int wr0 = tid >> 3, wq0 = (tid & 7) << 2;      // W: 2 float4/thread
    const int wr1 = (tid + 256) >> 3, wq1 = wq0;

    const int nchunk = K >> 5;

    // prologue: stage chunk 0 into buffer 0
    cp_async_b128(&X[(size_t)(rowBase + xr) * K + xq], &Xs[0][xr * GP + xq]);
    cp_async_b128(&W[(size_t)(colBase + wr0) * K + wq0], &Ws[0][wr0 * GP + wq0]);
    cp_async_b128(&W[(size_t)(colBase + wr1) * K + wq1], &Ws[0][wr1 * GP + wq1]);

    v8f c = {0.f, 0.f, 0.f, 0.f, 0.f, 0.f, 0.f, 0.f};

    for (int i = 0; i < nchunk; ++i) {
        cp_async_wait();                 // this wave's copies for chunk i done
        __syncthreads();                 // => whole block's chunk i visible

        if (i + 1 < nchunk) {            // issue chunk i+1 into other buffer
            const int kk = (i + 1) << 5;
            const int nb = (i + 1) & 1;
            cp_async_b128(&X[(size_t)(rowBase + xr) * K + kk + xq],
                          &Xs[nb][xr * GP + xq]);
            cp_async_b128(&W[(size_t)(colBase + wr0) * K + kk + wq0],
                          &Ws[nb][wr0 * GP + wq0]);
            cp_async_b128(&W[(size_t)(colBase + wr1) * K + kk + wq1],
                          &Ws[nb][wr1 * GP + wq1]);
        }

        const float* xs = Xs[i & 1];
        const float* ws = Ws[i & 1];
#pragma unroll
        for (int k4 = 0; k4 < 32; k4 += 4) {
            v2f a, b;
            a.x = xs[(tr + m) * GP + k4 + kb];
            a.y = xs[(tr + m) * GP + k4 + kb + 1];
            b.x = ws[(tc + m) * GP + k4 + kb];
            b.y = ws[(tc + m) * GP + k4 + kb + 1];
            c = wmma_f32(a, b, c);
        }
    }

    const int mo = (lane < 16) ? 0 : 8;  // C/D layout: VGPR r -> M=r or r+8
#pragma unroll
    for (int r = 0; r < 8; ++r) {
        int row = rowBase + tr + r + mo;
        int col = colBase + tc + m;
        Y[(size_t)row * M + col] = c[r] + bias[col];
    }
}

// ---------------------------------------------------------------------------
// Attention: one block per (query-tile qt, head h, batch b).
// Full score row S[16][2048] lives in LDS (~165 KB total, CDNA5 WGP = 320 KB)
// so softmax is exact and attn_weights is written to HBM exactly once, with
// b128 stores (each wave stores 512 contiguous bytes per instruction).
// grid: (T/16, H, B), block = 256 threads (8 waves).
// ---------------------------------------------------------------------------
#define SPITCH 2052                       // 2048 + 4 pad (16B-aligned rows)
__global__ __launch_bounds__(256)
void attn_kernel(const float* __restrict__ qkv,   // [B*T, 3072]
                 float* __restrict__ attnw,        // [B,H,T,T]
                 float* __restrict__ attno) {      // [B*T, 1024]
    extern __shared__ float smem[];
    float* S      = smem;                       // 16 * SPITCH
    float* Qs     = S + 16 * SPITCH;            // 16 * 68
    float* red    = Qs + 16 * 68;               // 16 * 16
    float* rowmax = red + 256;                  // 16
    float* rowinv = rowmax + 16;                // 16
    float* Op     = rowinv + 16;                // 8 * 16 * 64

    const int qt  = blockIdx.x;
    const int h   = blockIdx.y;
    const int b   = blockIdx.z;
    const int q0  = qt * 16;
    const int nkt = qt + 1;                     // causal: key tiles 0..qt
    const int ncols = nkt * 16;

    const int tid  = threadIdx.x;
    const int lane = tid & 31;
    const int wave = tid >> 5;
    const int m    = lane & 15;
    const int kb   = (lane < 16) ? 0 : 2;
    const int mo   = (lane < 16) ? 0 : 8;
    const float scale = 0.125f;                 // 1/sqrt(64)

    const size_t rs = (size_t)QKV_M;            // qkv row stride (floats)
    const float* Qp = qkv + (size_t)(b * T_SEQ) * rs + (size_t)h * D_HEAD;
    const float* Kp = Qp + D_MODEL;
    const float* Vp = Qp + 2 * D_MODEL;

    // ---- stage Q tile (16x64 = 256 float4, one async per thread) ----------
    {
        int r = tid >> 4, d4 = (tid & 15) << 2;
        cp_async_b128(&Qp[(size_t)(q0 + r) * rs + d4], &Qs[r * 68 + d4]);
        cp_async_wait();
    }
    __syncthreads();

    // ---- S = scale * Q K^T with causal mask (WMMA f32) --------------------
    for (int kt = wave; kt < nkt; kt += 8) {
        const int j0 = kt * 16;
        v8f c = {0.f, 0.f, 0.f, 0.f, 0.f, 0.f, 0.f, 0.f};
        const float* Krow = Kp + (size_t)(j0 + m) * rs;   // lane's key row
#pragma unroll
        for (int k4 = 0; k4 < D_HEAD; k4 += 4) {
            v2f a, bb;
            a.x  = Qs[m * 68 + k4 + kb];
            a.y  = Qs[m * 68 + k4 + kb + 1];
            bb.x = Krow[k4 + kb];
            bb.y = Krow[k4 + kb + 1];
            c = wmma_f32(a, bb, c);
        }
#pragma unroll
        for (int r = 0; r < 8; ++r) {
            int qrow = r + mo;
            int kcol = j0 + m;
            float s = c[r] * scale;
            S[qrow * SPITCH + kcol] =
                ((q0 + qrow) >= kcol) ? s : -__builtin_inff();
        }
    }
    __syncthreads();

    // ---- exact softmax over LDS row (float4 scans) ------------------------
    const int srow = tid & 15;
    const int seg  = tid >> 4;
    float mx = -__builtin_inff();
    for (int col4 = seg << 2; col4 < ncols; col4 += 64) {
        float4 v = *(const float4*)&S[srow * SPITCH + col4];
        mx = fmaxf(mx, fmaxf(fmaxf(v.x, v.y), fmaxf(v.z, v.w)));
    }
    red[srow * 16 + seg] = mx;
    __syncthreads();
    if (tid < 16) {
        float v = red[tid * 16];
#pragma unroll
        for (int i = 1; i < 16; ++i) v = fmaxf(v, red[tid * 16 + i]);
        rowmax[tid] = v;
    }
    __syncthreads();
    float rm = rowmax[srow];
    float sum = 0.f;
    for (int col4 = seg << 2; col4 < ncols; col4 += 64) {
        float4 v = *(const float4*)&S[srow * SPITCH + col4];
        sum += __expf(v.x - rm) + __expf(v.y - rm) +
               __expf(v.z - rm) + __expf(v.w - rm);
    }
    red[srow * 16 + seg] = sum;
    __syncthreads();
    if (tid < 16) {
        float v = 0.f;
#pragma unroll
        for (int i = 0; i < 16; ++i) v += red[tid * 16 + i];
        rowinv[tid] = 1.0f / v;
    }
    __syncthreads();

    // ---- normalize in LDS + single b128-coalesced write of attn_weights ---
    float* awbase = attnw + ((size_t)((b * N_HEADS + h) * T_SEQ + q0) << 11);
    for (int idx = tid; idx < 16 * (T_SEQ / 4); idx += 256) {
        int r  = idx >> 9;              // 512 float4 per row
        int c4 = (idx & 511) << 2;
        float4 pv = {0.f, 0.f, 0.f, 0.f};
        if (c4 < ncols) {               // ncols % 16 == 0 -> uniform per group
            float4 sv = *(const float4*)&S[r * SPITCH + c4];
            float rmr = rowmax[r], inv = rowinv[r];
            pv.x = __expf(sv.x - rmr) * inv;
            pv.y = __expf(sv.y - rmr) * inv;
            pv.z = __expf(sv.z - rmr) * inv;
            pv.w = __expf(sv.w - rmr) * inv;
            *(float4*)&S[r * SPITCH + c4] = pv;
        }
        *(float4*)&awbase[((size_t)r << 11) + c4] = pv;
    }
    __syncthreads();

    // ---- O = P @ V (WMMA f32), cross-wave partial reduce in LDS -----------
    v8f o[4];
#pragma unroll
    for (int t = 0; t < 4; ++t) o[t] = (v8f){0.f,0.f,0.f,0.f,0.f,0.f,0.f,0.f};

    for (int kt = wave; kt < nkt; kt += 8) {
        const int j0 = kt * 16;
#pragma unroll
        for (int k4 = 0; k4 < 16; k4 += 4) {
            v2f a;
            a.x = S[m * SPITCH + j0 + k4 + kb];
            a.y = S[m * SPITCH + j0 + k4 + kb + 1];
            const float* v0 = Vp + (size_t)(j0 + k4 + kb) * rs;
            const float* v1 = v0 + rs;
#pragma unroll
            for (int t = 0; t < 4; ++t) {
                v2f bb;
                bb.x = v0[t * 16 + m];
                bb.y = v1[t * 16 + m];
                o[t] = wmma_f32(a, bb, o[t]);
            }
        }
    }
#pragma unroll
    for (int r = 0; r < 8; ++r) {
#pragma unroll
        for (int t = 0; t < 4; ++t)
            Op[(wave * 16 + r + mo) * 64 + t * 16 + m] = o[t][r];
    }
    __syncthreads();

    for (int idx = tid; idx < 16 * 64; idx += 256) {
        int r = idx >> 6, d = idx & 63;
        float acc = 0.f;
#pragma unroll
        for (int w = 0; w < 8; ++w) acc += Op[(w * 16 + r) * 64 + d];
        attno[(size_t)(b * T_SEQ + q0 + r) * D_MODEL + h * D_HEAD + d] = acc;
    }
}

// ---------------------------------------------------------------------------
extern "C" void kernel_launch(void* const* d_in, const int* in_sizes, int n_in,
                              void* d_out, int out_size, void* d_ws,
                              size_t ws_size, hipStream_t stream) {
    (void)in_sizes; (void)n_in; (void)out_size; (void)ws_size;
    const float* x     = (const float*)d_in[0];
    const float* qkv_w = (const float*)d_in[1];
    const float* qkv_b = (const float*)d_in[2];
    const float* out_w = (const float*)d_in[3];
    const float* out_b = (const float*)d_in[4];

    float* out   = (float*)d_out;                         // [B,T,C]
    float* attnw = out + (size_t)ROWS * D_MODEL;          // [B,H,T,T]

    float* qkv   = (float*)d_ws;                          // [8192,3072] 96 MB
    float* attno = qkv + (size_t)ROWS * QKV_M;            // [8192,1024] 32 MB

    dim3 blk(256);

    // 1) qkv = x @ qkv_w^T + qkv_b
    dim3 g1(QKV_M / 64, ROWS / 32);
    gemm_xwt_bias_kernel<<<g1, blk, 0, stream>>>(x, qkv_w, qkv_b, qkv,
                                                 D_MODEL, QKV_M);

    // 2) attention (scores + softmax + attn_weights write + P@V)
    dim3 g2(T_SEQ / 16, N_HEADS, B_SZ);
    size_t shmem = (size_t)(16 * SPITCH + 16 * 68 + 256 + 16 + 16 +
                            8 * 16 * 64) * sizeof(float);
    attn_kernel<<<g2, blk, shmem, stream>>>(qkv, attnw, attno);

    // 3) out = attno @ out_w^T + out_b
    dim3 g3(D_MODEL / 64, ROWS / 32);
    gemm_xwt_bias_kernel<<<g3, blk, 0, stream>>>(attno, out_w, out_b, out,
                                                 D_MODEL, D_MODEL);
}